// Circuit_10857677325161
// MI455X (gfx1250) — compile-verified
//
#include <hip/hip_runtime.h>

// ---------------------------------------------------------------------------
// Circuit scan for MI455X (gfx1250, wave32).
//   Pass 1: feedforward terms  F[t] = [Wffpv@I[t] (2) ; Wffy@I[t] (3)]
//           as a (Tx6)x(6x5) GEMM via V_WMMA_F32_16X16X4_F32 (full f32).
//   Pass 2: contractive recurrence parallelized per-lane with burn-in chunks.
// ---------------------------------------------------------------------------

typedef __attribute__((ext_vector_type(2))) float v2f;
typedef __attribute__((ext_vector_type(8))) float v8f;

#define A_PV  0.25f
#define A_PYR 0.10f

__device__ __forceinline__ float reluf(float x) { return fmaxf(x, 0.0f); }

// Masked + rectified feed-forward weight for combined neuron n (0..4), input k (0..5).
// n=0,1 -> Wffpv rows with MASK_FFPV [[1,1,1,0,0,0],[0,0,0,1,1,1]]
// n=2..4 -> Wffy rows with MASK_FFY (row r keeps k/2 == r)
__device__ __forceinline__ float ff_weight(int n, int k,
                                           const float* __restrict__ Wffpv,
                                           const float* __restrict__ Wffy) {
  if (k >= 6 || n >= 5) return 0.0f;
  if (n < 2) {
    bool m = (n == 0) ? (k < 3) : (k >= 3);
    return m ? reluf(Wffpv[n * 6 + k]) : 0.0f;
  }
  int r = n - 2;
  return ((k >> 1) == r) ? reluf(Wffy[r * 6 + k]) : 0.0f;
}

// -------------------------- Pass 1: WMMA feedforward ------------------------
// Tile = 16 consecutive timesteps. A(16x4 f32): lane L holds M=L%16,
// K = 2*(L/16)+{0,1} in its two VGPRs. B(4x16): lane L holds N=L%16,
// K = 2*(L/16)+{0,1}. Two WMMAs cover K=0..7 (cols 6,7 zero-padded).
// D(16x16 f32): lane L, VGPR v -> (M = v + 8*(L/16), N = L%16).
__global__ void ff_wmma_kernel(const float* __restrict__ I,
                               const float* __restrict__ Wffpv,
                               const float* __restrict__ Wffy,
                               float* __restrict__ F,
                               int nTiles) {
  const int lane   = threadIdx.x & 31;
  const int wave   = (int)((blockIdx.x * blockDim.x + threadIdx.x) >> 5);
  const int nWaves = (int)((gridDim.x * blockDim.x) >> 5);
  const int n  = lane & 15;        // output-neuron column (also A row index m)
  const int kh = (lane >> 4) * 2;  // 0 or 2

  // B operand: rectified+masked feedforward weights, constant over the loop.
  v2f b0, b1;
  b0.x = ff_weight(n, kh + 0, Wffpv, Wffy);
  b0.y = ff_weight(n, kh + 1, Wffpv, Wffy);
  b1.x = ff_weight(n, 4 + kh + 0, Wffpv, Wffy);
  b1.y = ff_weight(n, 4 + kh + 1, Wffpv, Wffy);

  const float khide = (kh == 0) ? 1.0f : 0.0f;  // zero-pad A cols 6,7 without EXEC games

  for (int tile = wave; tile < nTiles; tile += nWaves) {
    const long t0 = (long)tile * 16;
    const float* rowp = I + (t0 + n) * 6;        // this lane's timestep row

    v2f a0 = *(const v2f*)(rowp + kh);           // K = kh, kh+1   (8B aligned)
    v2f a1 = *(const v2f*)(rowp + 4);            // K = 4,5 for kh==0
    a1.x *= khide;                               // K = 6,7 -> 0 for kh==2
    a1.y *= khide;

    v8f acc = {};
    acc = __builtin_amdgcn_wmma_f32_16x16x4_f32(false, a0, false, b0,
                                                (short)0, acc, false, false);
    acc = __builtin_amdgcn_wmma_f32_16x16x4_f32(false, a1, false, b1,
                                                (short)0, acc, false, false);

    if (n < 5) {                                 // only 5 real output columns
      const int mbase = (lane >> 4) * 8;
      float* dst = F + (t0 + mbase) * 8 + n;     // padded [t][8] layout
#pragma unroll
      for (int v = 0; v < 8; ++v) dst[(long)v * 8] = acc[v];
    }
  }
}

// ---------------------- Pass 2: chunked recurrence scan ---------------------
// One lane per output chunk of CHUNK steps, warmed up from zero state BURN
// steps earlier (EMA contraction: worst-case residual 0.989^BURN ~ 4e-8).
template <bool USE_F>
__global__ void scan_kernel(const float* __restrict__ I,
                            const float* __restrict__ F,
                            const float* __restrict__ Wffpv,
                            const float* __restrict__ Wlat,
                            const float* __restrict__ Wffy,
                            const float* __restrict__ Wiy,
                            const float* __restrict__ Wffh,
                            const float* __restrict__ Wfby,
                            float* __restrict__ out,
                            int T, int nChunks, int chunkLen, int burn) {
  const int c = (int)(blockIdx.x * blockDim.x + threadIdx.x);
  if (c >= nChunks) return;

  // Recurrent weights: rectified and masked once, held in registers.
  const float wl00 = reluf(Wlat[0]), wl01 = reluf(Wlat[1]);   // MASK_LAT [[1,1,0],[0,1,1]]
  const float wl11 = reluf(Wlat[4]), wl12 = reluf(Wlat[5]);
  const float wi00 = reluf(Wiy[0]);                            // MASK_IY [[1,0],[1,1],[0,1]]
  const float wi10 = reluf(Wiy[2]), wi11 = reluf(Wiy[3]);
  const float wi21 = reluf(Wiy[5]);
  const float wb00 = reluf(Wfby[0]), wb01 = reluf(Wfby[1]);    // full 3x2
  const float wb10 = reluf(Wfby[2]), wb11 = reluf(Wfby[3]);
  const float wb20 = reluf(Wfby[4]), wb21 = reluf(Wfby[5]);
  const float wh00 = reluf(Wffh[0]), wh01 = reluf(Wffh[1]), wh02 = reluf(Wffh[2]);
  const float wh10 = reluf(Wffh[3]), wh11 = reluf(Wffh[4]), wh12 = reluf(Wffh[5]);

  // Inline feedforward weights (fallback when workspace too small for F).
  float wp00 = 0, wp01 = 0, wp02 = 0, wp13 = 0, wp14 = 0, wp15 = 0;
  float wy00 = 0, wy01 = 0, wy12 = 0, wy13 = 0, wy24 = 0, wy25 = 0;
  if (!USE_F) {
    wp00 = reluf(Wffpv[0]);  wp01 = reluf(Wffpv[1]);  wp02 = reluf(Wffpv[2]);
    wp13 = reluf(Wffpv[9]);  wp14 = reluf(Wffpv[10]); wp15 = reluf(Wffpv[11]);
    wy00 = reluf(Wffy[0]);   wy01 = reluf(Wffy[1]);
    wy12 = reluf(Wffy[8]);   wy13 = reluf(Wffy[9]);
    wy24 = reluf(Wffy[16]);  wy25 = reluf(Wffy[17]);
  }

  const long Tl    = (long)T;
  const long tPay0 = (long)c * chunkLen;       // first recorded step
  long tPay1 = tPay0 + chunkLen; if (tPay1 > Tl) tPay1 = Tl;
  long t = tPay0 - burn; if (t < 0) t = 0;     // chunks 0/1 start exactly at 0

  float pyr0 = 0, pyr1 = 0, pyr2 = 0, pv0 = 0, pv1 = 0, hva0 = 0, hva1 = 0;

  for (; t < tPay1; ++t) {
    // ---- feedforward drive for this step ----
    float fpv0, fpv1, fpy0, fpy1, fpy2;
    if (USE_F) {
      const float4 f4 = *(const float4*)(F + t * 8);   // 32B/step, 16B aligned
      fpv0 = f4.x; fpv1 = f4.y; fpy0 = f4.z; fpy1 = f4.w;
      fpy2 = F[t * 8 + 4];
      __builtin_prefetch(F + (t + 64) * 8, 0, 1);      // ~2KB ahead per lane
    } else {
      const v2f iA = *(const v2f*)(I + t * 6);
      const v2f iB = *(const v2f*)(I + t * 6 + 2);
      const v2f iC = *(const v2f*)(I + t * 6 + 4);
      fpv0 = wp00 * iA.x + wp01 * iA.y + wp02 * iB.x;
      fpv1 = wp13 * iB.y + wp14 * iC.x + wp15 * iC.y;
      fpy0 = wy00 * iA.x + wy01 * iA.y;
      fpy1 = wy12 * iB.x + wy13 * iB.y;
      fpy2 = wy24 * iC.x + wy25 * iC.y;
    }

    // ---- PV stage ----
    const float upv0 = fpv0 + wl00 * pyr0 + wl01 * pyr1;
    const float upv1 = fpv1 + wl11 * pyr1 + wl12 * pyr2;
    pv0 = A_PV * reluf(upv0) + (1.0f - A_PV) * pv0;
    pv1 = A_PV * reluf(upv1) + (1.0f - A_PV) * pv1;

    // ---- Pyramidal stage ----
    const float upy0 = fpy0 - wi00 * pv0 + wb00 * hva0 + wb01 * hva1;
    const float upy1 = fpy1 - wi10 * pv0 - wi11 * pv1 + wb10 * hva0 + wb11 * hva1;
    const float upy2 = fpy2 - wi21 * pv1 + wb20 * hva0 + wb21 * hva1;
    const float npy0 = A_PYR * reluf(upy0) + (1.0f - A_PYR) * pyr0;
    const float npy1 = A_PYR * reluf(upy1) + (1.0f - A_PYR) * pyr1;
    const float npy2 = A_PYR * reluf(upy2) + (1.0f - A_PYR) * pyr2;

    // ---- record (pyr/pv post-update, hva PRE-update), layout (7,T) ----
    if (t >= tPay0) {
      out[0 * Tl + t] = npy0;
      out[1 * Tl + t] = npy1;
      out[2 * Tl + t] = npy2;
      out[3 * Tl + t] = pv0;
      out[4 * Tl + t] = pv1;
      out[5 * Tl + t] = hva0;
      out[6 * Tl + t] = hva1;
    }

    // ---- HVA stage (off the pyr->pv critical path) ----
    const float uh0 = wh00 * npy0 + wh01 * npy1 + wh02 * npy2;
    const float uh1 = wh10 * npy0 + wh11 * npy1 + wh12 * npy2;
    hva0 = A_PYR * reluf(uh0) + (1.0f - A_PYR) * hva0;
    hva1 = A_PYR * reluf(uh1) + (1.0f - A_PYR) * hva1;
    pyr0 = npy0; pyr1 = npy1; pyr2 = npy2;
  }
}

// ------------------------------- launcher ----------------------------------
extern "C" void kernel_launch(void* const* d_in, const int* in_sizes, int n_in,
                              void* d_out, int out_size, void* d_ws, size_t ws_size,
                              hipStream_t stream) {
  const float* I     = (const float*)d_in[0];
  const float* Wffpv = (const float*)d_in[1];
  const float* Wlat  = (const float*)d_in[2];
  const float* Wffy  = (const float*)d_in[3];
  const float* Wiy   = (const float*)d_in[4];
  const float* Wffh  = (const float*)d_in[5];
  const float* Wfby  = (const float*)d_in[6];
  float* out = (float*)d_out;

  const int T = in_sizes[0] / 6;

  const int CHUNK = 1024;   // recorded steps per lane
  const int BURN  = 1536;   // warm-up steps (worst-case residual ~4e-8)
  const int nChunks = (T + CHUNK - 1) / CHUNK;
  const int blocks  = (nChunks + 255) / 256;

  const size_t needF = (size_t)T * 8 * sizeof(float);
  float* F = (float*)d_ws;
  const bool useF = (ws_size >= needF) && (T % 16 == 0);

  if (useF) {
    const int nTiles = T / 16;
    ff_wmma_kernel<<<512, 256, 0, stream>>>(I, Wffpv, Wffy, F, nTiles);
    scan_kernel<true><<<blocks, 256, 0, stream>>>(I, F, Wffpv, Wlat, Wffy, Wiy,
                                                  Wffh, Wfby, out, T, nChunks,
                                                  CHUNK, BURN);
  } else {
    scan_kernel<false><<<blocks, 256, 0, stream>>>(I, nullptr, Wffpv, Wlat, Wffy,
                                                   Wiy, Wffh, Wfby, out, T,
                                                   nChunks, CHUNK, BURN);
  }
}